// InternalGraphConvolutionLayer_36112085025448
// MI455X (gfx1250) — compile-verified
//
#include <hip/hip_runtime.h>
#include <hip/hip_bf16.h>
#include <math.h>

typedef float v2f __attribute__((ext_vector_type(2)));
typedef float v4f __attribute__((ext_vector_type(4)));
typedef float v8f __attribute__((ext_vector_type(8)));

#define N_NODES_C 50000
#define D_FEAT    128
#define N_TILES   (N_NODES_C / 16)   // 3125

// ---------------------------------------------------------------------------
// ws layout: [0,512) = result accumulator (128 floats, padded)
//            [512, 512 + N*128*4) = per-node neighbor sums (nsum)
// ---------------------------------------------------------------------------

__global__ void k_zero(float* __restrict__ p, int n4) {
    int i = blockIdx.x * blockDim.x + threadIdx.x;
    if (i < n4) ((v4f*)p)[i] = (v4f){0.f, 0.f, 0.f, 0.f};
}

// Segment-sum over sorted segments: each wave owns a contiguous edge chunk,
// accumulates runs of equal segment ids in registers, flushes with f32 atomics.
__global__ void k_segsum(const float* __restrict__ impact,
                         const int*  __restrict__ nbr_ids,
                         const int*  __restrict__ nbr_segs,
                         float* __restrict__ nsum,
                         int E, int edges_per_wave) {
    int wave = blockIdx.x * (blockDim.x >> 5) + (threadIdx.x >> 5);
    int lane = threadIdx.x & 31;
    long start = (long)wave * edges_per_wave;
    long end   = start + edges_per_wave;
    if (end > E) end = E;
    if (start >= E) return;

    v4f acc = (v4f){0.f, 0.f, 0.f, 0.f};
    int cur = -1;
    for (long e = start; e < end; ++e) {
        int seg = nbr_segs[e];
        int rid = nbr_ids[e];
        if (seg != cur) {
            if (cur >= 0) {
                float* dst = nsum + (size_t)cur * D_FEAT + lane * 4;
                atomicAdd(dst + 0, acc.x); atomicAdd(dst + 1, acc.y);
                atomicAdd(dst + 2, acc.z); atomicAdd(dst + 3, acc.w);
            }
            acc = (v4f){0.f, 0.f, 0.f, 0.f};
            cur = seg;
        }
        const v4f v = *(const v4f*)(impact + (size_t)rid * D_FEAT + lane * 4);
        acc += v;
    }
    if (cur >= 0) {
        float* dst = nsum + (size_t)cur * D_FEAT + lane * 4;
        atomicAdd(dst + 0, acc.x); atomicAdd(dst + 1, acc.y);
        atomicAdd(dst + 2, acc.z); atomicAdd(dst + 3, acc.w);
    }
}

// Fused GEMM (K=256: [self | nsum] @ [W^T ; M^T]) + relu + column-sum.
// 8 waves/block; wave w owns output columns [16w, 16w+16).
// A tiles staged in LDS then prefetched whole into VGPRs; B pinned in VGPRs.
__global__ void __launch_bounds__(256)
k_gemm_relu_sum(const float* __restrict__ impact,
                const float* __restrict__ W,
                const float* __restrict__ Mm,
                const int*  __restrict__ node_ids,
                const float* __restrict__ nsum,
                float* __restrict__ res_acc) {
    __shared__ v2f A2[128 * 16];       // [k_pair 0..127][row 0..15], 16 KB

    const int tid  = threadIdx.x;
    const int wave = tid >> 5;          // n-tile owned by this wave
    const int lane = tid & 31;
    const int r    = lane & 15;         // row (A/M index) or col (B/N index)
    const int h    = lane >> 4;         // K-half selector
    const int n    = wave * 16 + r;     // this lane's output column

    // Preload B = [W^T ; M^T] in WMMA 4x16 operand layout:
    // breg[kk] = { B[4kk+2h][n], B[4kk+2h+1][n] } with B[k][n] = W[n][k] / M[n][k-128]
    v2f breg[64];
    #pragma unroll
    for (int kk = 0; kk < 32; ++kk)
        breg[kk]      = *(const v2f*)(W  + (size_t)n * D_FEAT + 4 * kk + 2 * h);
    #pragma unroll
    for (int kk = 0; kk < 32; ++kk)
        breg[32 + kk] = *(const v2f*)(Mm + (size_t)n * D_FEAT + 4 * kk + 2 * h);

    const int row  = tid >> 3;          // 0..31 : 16 self rows + 16 nsum rows
    const int part = tid & 7;
    const int rr   = row & 15;
    float S = 0.f;                      // running relu'd partial (per lane)

    for (int t = blockIdx.x; t < N_TILES; t += gridDim.x) {
        __syncthreads();
        // Stage A tile: rows 0-15 = gathered self feats, rows 16-31 = nsum rows
        const float* src;
        int slotbase;
        if (row < 16) { src = impact + (size_t)node_ids[t * 16 + row] * D_FEAT; slotbase = 0; }
        else          { src = nsum   + (size_t)(t * 16 + rr) * D_FEAT;          slotbase = 64; }
        #pragma unroll
        for (int i = 0; i < 8; ++i) {
            int k2 = part + 8 * i;
            A2[(slotbase + k2) * 16 + rr] = *(const v2f*)(src + 2 * k2);
        }
        __syncthreads();

        // Prefetch the whole A slice for this lane into VGPRs so the DS loads
        // issue as one batch and the 64 WMMAs run back-to-back (no per-pair
        // s_wait_dscnt stalls on the matrix pipe).
        v2f areg[64];
        #pragma unroll
        for (int kk = 0; kk < 64; ++kk)
            areg[kk] = A2[(2 * kk + h) * 16 + r];

        v8f acc = {};
        #pragma unroll
        for (int kk = 0; kk < 64; ++kk)
            acc = __builtin_amdgcn_wmma_f32_16x16x4_f32(
                      false, areg[kk], false, breg[kk], (short)0, acc, false, false);

        #pragma unroll
        for (int v = 0; v < 8; ++v) S += fmaxf(acc[v], 0.f);
    }

    // Column N lives in lanes N and N+16 (C/D layout): fold halves, then commit.
    float tot = S + __shfl_xor(S, 16, 32);
    if (lane < 16) atomicAdd(&res_acc[wave * 16 + lane], tot);
}

__global__ void k_softmax128(const float* __restrict__ acc, float* __restrict__ out) {
    __shared__ float sh[128];
    int t = threadIdx.x;
    float v = acc[t];
    sh[t] = v;
    __syncthreads();
    for (int s = 64; s > 0; s >>= 1) { if (t < s) sh[t] = fmaxf(sh[t], sh[t + s]); __syncthreads(); }
    float m = sh[0];
    __syncthreads();
    float e = expf(v - m);
    sh[t] = e;
    __syncthreads();
    for (int s = 64; s > 0; s >>= 1) { if (t < s) sh[t] += sh[t + s]; __syncthreads(); }
    out[t] = e / sh[0];
}

extern "C" void kernel_launch(void* const* d_in, const int* in_sizes, int n_in,
                              void* d_out, int out_size, void* d_ws, size_t ws_size,
                              hipStream_t stream) {
    const float* impact   = (const float*)d_in[0];
    const float* W        = (const float*)d_in[1];
    const float* Mm       = (const float*)d_in[2];
    const int*   node_ids = (const int*)d_in[3];
    const int*   nbr_ids  = (const int*)d_in[4];
    const int*   nbr_segs = (const int*)d_in[5];
    float* out = (float*)d_out;

    float* res_acc = (float*)d_ws;
    float* nsum    = res_acc + 128;      // 512-byte offset

    const int E = in_sizes[4];

    // 1) zero result accumulator + nsum (contiguous region)
    int total4 = (128 + N_NODES_C * D_FEAT) / 4;
    k_zero<<<(total4 + 255) / 256, 256, 0, stream>>>((float*)d_ws, total4);

    // 2) sorted-segment sum with run-length accumulation
    const int epw = 256;
    int waves  = (E + epw - 1) / epw;
    int blocks = (waves + 7) / 8;
    k_segsum<<<blocks, 256, 0, stream>>>(impact, nbr_ids, nbr_segs, nsum, E, epw);

    // 3) fused fp32 WMMA GEMM + relu + column reduction
    k_gemm_relu_sum<<<256, 256, 0, stream>>>(impact, W, Mm, node_ids, nsum, res_acc);

    // 4) softmax over D=128
    k_softmax128<<<1, 128, 0, stream>>>(res_acc, out);
}